// TokenRouter_65687229825450
// MI455X (gfx1250) — compile-verified
//
#include <hip/hip_runtime.h>

#define B_   8
#define T_   8192
#define D_   4096
#define NTOK (B_ * T_)   // 65536 tokens
#define KSEL (T_ / 2)    // 4096 = k for top-k

typedef __attribute__((ext_vector_type(2))) float v2f;
typedef __attribute__((ext_vector_type(8))) float v8f;

// -------- Kernel A: streaming fp32 GEMV, WMMA f32 16x16x4 cross-lane reduce -----
// Each wave owns 16 consecutive tokens. Inner loop: lane loads one float4 of w
// (L0/L2-resident, 16 KB total) and 16 float4s of x (fully coalesced 512B/wave
// per instruction), fp32 FMA into 16 per-lane accumulators. A small LDS
// transpose (stride 33 -> provably bank-conflict-free) reshapes the 16x32
// partial-sum matrix into an A-fragment of V_WMMA_F32_16X16X4_F32; one WMMA
// against an all-ones B matrix yields all 16 logits (row sums) exactly in fp32.
__global__ __launch_bounds__(256) void router_logits_kernel(
    const float* __restrict__ x, const float* __restrict__ w,
    float* __restrict__ logits) {
  __shared__ float P[8][16 * 33];  // per-wave 16x32 partials, padded stride 33

  const int lane = threadIdx.x & 31;
  const int wave = threadIdx.x >> 5;
  const int t0 = (blockIdx.x * 8 + wave) * 16;

  const float4* __restrict__ xv = (const float4*)x;
  const float4* __restrict__ wv = (const float4*)w;

  float acc[16];
#pragma unroll
  for (int i = 0; i < 16; ++i) acc[i] = 0.0f;

  // D/4 = 1024 float4 per token row; lane covers float4 index lane + 32*k
#pragma unroll 1
  for (int k = 0; k < 32; ++k) {
    const int fi = lane + (k << 5);
    const float4 wq = wv[fi];
#pragma unroll
    for (int tt = 0; tt < 16; ++tt) {
      const float4 xq = xv[(size_t)(t0 + tt) * (D_ / 4) + fi];
      acc[tt] = fmaf(xq.x, wq.x, acc[tt]);
      acc[tt] = fmaf(xq.y, wq.y, acc[tt]);
      acc[tt] = fmaf(xq.z, wq.z, acc[tt]);
      acc[tt] = fmaf(xq.w, wq.w, acc[tt]);
    }
  }

  // Transpose partials through LDS. For fixed tt, lanes write consecutive
  // addresses (conflict-free). Reads: addr = M*33 + g*16 + i; 33 is self-inverse
  // mod 64, so all 32 (M,g) combos hit distinct banks.
#pragma unroll
  for (int tt = 0; tt < 16; ++tt) P[wave][tt * 33 + lane] = acc[tt];
  __syncthreads();

  const int M = lane & 15;
  const int g = lane >> 4;  // K-half: lanes 0-15 -> K0/K1, lanes 16-31 -> K2/K3
  float a0 = 0.0f, a1 = 0.0f;
#pragma unroll
  for (int i = 0; i < 8; ++i) a0 += P[wave][M * 33 + g * 16 + i];
#pragma unroll
  for (int i = 0; i < 8; ++i) a1 += P[wave][M * 33 + g * 16 + 8 + i];

  // A-fragment (16x4 f32): lane L<16 holds [M=L, K0|K1], lane L>=16 [M=L-16, K2|K3]
  v2f Afrag;
  Afrag.x = a0;
  Afrag.y = a1;
  v2f Bones;
  Bones.x = 1.0f;
  Bones.y = 1.0f;
  v8f Cacc = {};
  // D[M][n] = sum_K A[M][K] * 1 = logit(token t0+M), replicated across n
  Cacc = __builtin_amdgcn_wmma_f32_16x16x4_f32(
      /*neg_a=*/false, Afrag, /*neg_b=*/false, Bones,
      /*c_mod=*/(short)0, Cacc, /*reuse_a=*/false, /*reuse_b=*/false);

  // C/D layout: VGPR r, lanes 0-15 -> M=r (N=lane), lanes 16-31 -> M=r+8
  if (lane == 0) {
#pragma unroll
    for (int i = 0; i < 8; ++i) logits[t0 + i] = Cacc[i];
  } else if (lane == 16) {
#pragma unroll
    for (int i = 0; i < 8; ++i) logits[t0 + 8 + i] = Cacc[i];
  }
}

// -------- Kernel B: per-row top-k (k = T/2) via 32-pass bitwise radix select ---
// One 1024-thread workgroup per batch row. Keys (order-preserving uint map of
// the fp32 logits) live in LDS (32 KB). tau = k-th largest key found by MSB->LSB
// prefix search with block-wide counting; mask = (key > tau) plus the first
// (k - count_gt) elements equal to tau.
__global__ __launch_bounds__(1024) void router_topk_kernel(
    const float* __restrict__ logits, float* __restrict__ mask) {
  __shared__ unsigned keys[T_];
  __shared__ int cnts[40];  // [0..31] per-bit counts, [32] count>tau, [33] eq ticket

  const int tid = threadIdx.x;
  const int row = blockIdx.x;
  const float* __restrict__ lrow = logits + (size_t)row * T_;

#pragma unroll
  for (int i = 0; i < 8; ++i) {
    const int idx = tid + i * 1024;
    unsigned u = __float_as_uint(lrow[idx]);
    u = (u & 0x80000000u) ? ~u : (u | 0x80000000u);  // monotone float->uint
    keys[idx] = u;
  }
  if (tid < 40) cnts[tid] = 0;
  __syncthreads();

  unsigned prefix = 0u;
#pragma unroll 1
  for (int bit = 31; bit >= 0; --bit) {
    const unsigned cand = prefix | (1u << bit);
    int c = 0;
#pragma unroll
    for (int i = 0; i < 8; ++i) c += (keys[tid + i * 1024] >= cand) ? 1 : 0;
#pragma unroll
    for (int off = 16; off > 0; off >>= 1) c += __shfl_xor(c, off, 32);
    if ((tid & 31) == 0) atomicAdd(&cnts[bit], c);
    __syncthreads();
    if (cnts[bit] >= KSEL) prefix = cand;  // uniform across block
  }
  const unsigned tau = prefix;  // k-th largest key

  int cg = 0;
#pragma unroll
  for (int i = 0; i < 8; ++i) cg += (keys[tid + i * 1024] > tau) ? 1 : 0;
#pragma unroll
  for (int off = 16; off > 0; off >>= 1) cg += __shfl_xor(cg, off, 32);
  if ((tid & 31) == 0) atomicAdd(&cnts[32], cg);
  __syncthreads();
  const int take = KSEL - cnts[32];  // ties at tau to accept

#pragma unroll
  for (int i = 0; i < 8; ++i) {
    const int idx = tid + i * 1024;
    const unsigned u = keys[idx];
    float m = 0.0f;
    if (u > tau) {
      m = 1.0f;
    } else if (u == tau) {
      const int p = atomicAdd(&cnts[33], 1);
      m = (p < take) ? 1.0f : 0.0f;
    }
    mask[(size_t)row * T_ + idx] = m;
  }
}

extern "C" void kernel_launch(void* const* d_in, const int* in_sizes, int n_in,
                              void* d_out, int out_size, void* d_ws,
                              size_t ws_size, hipStream_t stream) {
  (void)in_sizes; (void)n_in; (void)d_ws; (void)ws_size; (void)out_size;
  const float* x = (const float*)d_in[0];  // (B,T,D) fp32
  const float* w = (const float*)d_in[1];  // (D,)    fp32
  float* maskOut = (float*)d_out;          // first B*T: mask (as 0/1 floats)
  float* logitsOut = maskOut + NTOK;       // second B*T: logits fp32

  // 512 blocks x 256 threads; each block's 8 waves cover 8*16 = 128 tokens
  router_logits_kernel<<<NTOK / 128, 256, 0, stream>>>(x, w, logitsOut);
  // one workgroup per batch row; stream order makes logits visible
  router_topk_kernel<<<B_, 1024, 0, stream>>>(logitsOut, maskOut);
}